// RGCN_46497315946868
// MI455X (gfx1250) — compile-verified
//
#include <hip/hip_runtime.h>

typedef __attribute__((ext_vector_type(2))) float v2f;
typedef __attribute__((ext_vector_type(8))) float v8f;

#define N_NODES 50000
#define R_REL   16
#define H_DIM   32
#define L_DIM   8
#define E_EDGES 1600000

#define K_STEPS ((R_REL + 1) * (H_DIM / 4))   // 136 WMMA steps of K=4
#define TILES   (N_NODES / 16)                // 3125, exact

// ---------------- zero init ----------------
__global__ void rgcn_zero_f32(float* __restrict__ p, long long n) {
  long long i = (long long)blockIdx.x * blockDim.x + threadIdx.x;
  long long stride = (long long)gridDim.x * blockDim.x;
  for (; i < n; i += stride) p[i] = 0.0f;
}

// ---------------- per-(relation,dst) edge counts ----------------
__global__ void rgcn_count(const int* __restrict__ ei, const int* __restrict__ et,
                           float* __restrict__ cnt) {
  int e = blockIdx.x * blockDim.x + threadIdx.x;
  if (e >= E_EDGES) return;
  int dst = ei[E_EDGES + e];
  int r   = et[e];
  atomicAdd(&cnt[(size_t)r * N_NODES + dst], 1.0f);
}

// ---------------- layer 1 scatter: agg1[dst,h] += W1[r,src,h]/deg ----------------
__global__ void rgcn_scatter1(const int* __restrict__ ei, const int* __restrict__ et,
                              const float* __restrict__ cnt, const float* __restrict__ W1,
                              float* __restrict__ agg1) {
  long long t = (long long)blockIdx.x * blockDim.x + threadIdx.x;
  if (t >= (long long)E_EDGES * H_DIM) return;
  int e = (int)(t >> 5);      // H_DIM == 32 == lanes per wave
  int h = (int)(t & 31);
  int src = ei[e];
  int dst = ei[E_EDGES + e];
  int r   = et[e];
  float c = cnt[(size_t)r * N_NODES + dst];
  float norm = 1.0f / fmaxf(c, 1.0f);
  float w = W1[((size_t)r * N_NODES + src) * H_DIM + h];
  atomicAdd(&agg1[(size_t)dst * H_DIM + h], w * norm);
}

// ---------------- h1 = relu(agg1 + root1 + b1) ----------------
__global__ void rgcn_h1(const float* __restrict__ agg1, const float* __restrict__ root1,
                        const float* __restrict__ b1, float* __restrict__ h1) {
  int t = blockIdx.x * blockDim.x + threadIdx.x;
  if (t >= N_NODES * H_DIM) return;
  float v = agg1[t] + root1[t] + b1[t & (H_DIM - 1)];
  h1[t] = fmaxf(v, 0.0f);
}

// ---------------- layer 2 scatter: A[r,dst,h] += h1[src,h]/deg ----------------
__global__ void rgcn_scatter2(const int* __restrict__ ei, const int* __restrict__ et,
                              const float* __restrict__ cnt, const float* __restrict__ h1,
                              float* __restrict__ A) {
  long long t = (long long)blockIdx.x * blockDim.x + threadIdx.x;
  if (t >= (long long)E_EDGES * H_DIM) return;
  int e = (int)(t >> 5);
  int h = (int)(t & 31);
  int src = ei[e];
  int dst = ei[E_EDGES + e];
  int r   = et[e];
  float c = cnt[(size_t)r * N_NODES + dst];
  float norm = 1.0f / fmaxf(c, 1.0f);
  float v = h1[(size_t)src * H_DIM + h] * norm;
  atomicAdd(&A[((size_t)r * N_NODES + dst) * H_DIM + h], v);
}

// ---------------- output GEMM via V_WMMA_F32_16X16X4_F32 ----------------
// One wave per 16-node tile; 8 waves (8 tiles) per block. The padded B panel
// (W2 for r=0..15 then root2, cols 8..15 zero) is staged once per block into
// LDS, pre-swizzled into per-lane WMMA operand order so the hot loop is just
// ds_load_b64 + global_load_b64 + v_wmma per K=4 step. No exec divergence in
// the loop -> EXEC stays all-ones for WMMA.
__global__ void rgcn_out_wmma(const float* __restrict__ Amat, const float* __restrict__ h1,
                              const float* __restrict__ W2, const float* __restrict__ root2,
                              const float* __restrict__ b2, float* __restrict__ out) {
  __shared__ float ldsB[K_STEPS * 64];   // 34,816 bytes

  // Cooperative fill: entry (step s, lane, j) holds B[k][n] with
  // n = lane&15, k = (s&7)*4 + (lane>>4)*2 + j  (WMMA 16x16x4 B layout).
  for (int idx = threadIdx.x; idx < K_STEPS * 64; idx += blockDim.x) {
    int s    = idx >> 6;
    int slot = idx & 63;
    int ln   = slot >> 1;
    int j    = slot & 1;
    int n    = ln & 15;
    int k    = ((s & 7) << 2) + ((ln >> 4) << 1) + j;
    int rr   = s >> 3;
    float v = 0.0f;
    if (n < L_DIM) {
      const float* Bseg = (rr < R_REL) ? (W2 + (size_t)rr * H_DIM * L_DIM) : root2;
      v = Bseg[(size_t)k * L_DIM + n];
    }
    ldsB[idx] = v;
  }
  __syncthreads();

  int wave = (blockIdx.x * blockDim.x + threadIdx.x) >> 5;
  int lane = threadIdx.x & 31;
  if (wave >= TILES) return;            // uniform per wave -> EXEC all 1s below

  const int row0 = wave * 16;
  const int m    = lane & 15;           // A row owned by this lane
  const int koff = (lane >> 4) * 2;     // K sub-offset per half-wave

  v8f acc = {};
  for (int rr = 0; rr <= R_REL; ++rr) {
    const float* Aseg = (rr < R_REL) ? (Amat + (size_t)rr * N_NODES * H_DIM) : h1;
    const float* arow = Aseg + (size_t)(row0 + m) * H_DIM + koff;
    const float* brow = ldsB + (size_t)rr * (H_DIM / 4) * 64 + lane * 2;
    #pragma unroll
    for (int k0 = 0; k0 < H_DIM; k0 += 4) {
      v2f a = *(const v2f*)(arow + k0);
      v2f b = *(const v2f*)(brow + (k0 >> 2) * 64);
      acc = __builtin_amdgcn_wmma_f32_16x16x4_f32(
          /*neg_a=*/false, a, /*neg_b=*/false, b,
          /*c_mod=*/(short)0, acc, /*reuse_a=*/false, /*reuse_b=*/false);
    }
  }

  // D layout: VGPR i -> row (i + 8*(lane>>4)), col = lane&15
  int l = lane & 15;
  if (l < L_DIM) {
    float bias = b2[l];
    int half = lane >> 4;
    #pragma unroll
    for (int i = 0; i < 8; ++i) {
      int node = row0 + i + 8 * half;
      float x = acc[i] + bias;
      out[(size_t)node * L_DIM + l] = 1.0f / (1.0f + __expf(-x));
    }
  }
}

extern "C" void kernel_launch(void* const* d_in, const int* in_sizes, int n_in,
                              void* d_out, int out_size, void* d_ws, size_t ws_size,
                              hipStream_t stream) {
  const int*   ei    = (const int*)d_in[0];   // [2, E]
  const int*   et    = (const int*)d_in[1];   // [E]
  const float* W1    = (const float*)d_in[2]; // [R, N, H]
  const float* root1 = (const float*)d_in[3]; // [N, H]
  const float* b1    = (const float*)d_in[4]; // [H]
  const float* W2    = (const float*)d_in[5]; // [R, H, L]
  const float* root2 = (const float*)d_in[6]; // [H, L]
  const float* b2    = (const float*)d_in[7]; // [L]
  float*       out   = (float*)d_out;         // [N, L]

  // workspace layout (floats)
  float* ws   = (float*)d_ws;
  float* cnt  = ws;                                        // R*N
  float* agg1 = cnt + (size_t)R_REL * N_NODES;             // N*H
  float* Amat = agg1 + (size_t)N_NODES * H_DIM;            // R*N*H
  float* h1   = Amat + (size_t)R_REL * N_NODES * H_DIM;    // N*H

  const long long nZero = (long long)R_REL * N_NODES          // cnt
                        + (long long)N_NODES * H_DIM          // agg1
                        + (long long)R_REL * N_NODES * H_DIM; // Amat
  rgcn_zero_f32<<<2048, 256, 0, stream>>>(ws, nZero);

  rgcn_count<<<(E_EDGES + 255) / 256, 256, 0, stream>>>(ei, et, cnt);

  const long long eh = (long long)E_EDGES * H_DIM;
  const int ehBlocks = (int)((eh + 255) / 256);
  rgcn_scatter1<<<ehBlocks, 256, 0, stream>>>(ei, et, cnt, W1, agg1);

  rgcn_h1<<<(N_NODES * H_DIM + 255) / 256, 256, 0, stream>>>(agg1, root1, b1, h1);

  rgcn_scatter2<<<ehBlocks, 256, 0, stream>>>(ei, et, cnt, h1, Amat);

  const int gemmBlocks = (TILES * 32 + 255) / 256;           // 8 tiles per block
  rgcn_out_wmma<<<gemmBlocks, 256, 0, stream>>>(Amat, h1, W2, root2, b2, out);

  (void)in_sizes; (void)n_in; (void)out_size; (void)ws_size;
}